// EVGNNBackbone_63385127354597
// MI455X (gfx1250) — compile-verified
//
#include <hip/hip_runtime.h>

typedef __attribute__((ext_vector_type(2))) float v2f;
typedef __attribute__((ext_vector_type(8))) float v8f;

#define WPB 4            // waves per block (blockDim = 128)
#define MAX_CIN 130
#define MAX_K (8 * MAX_CIN)

// ---------------- build hin for layer 1: concat(x, pos.xy) ----------------
__global__ void k_build_hin0(const float* __restrict__ x, const float* __restrict__ pos,
                             float* __restrict__ hin, int V) {
    int v = blockIdx.x * blockDim.x + threadIdx.x;
    if (v < V) {
        hin[v * 3 + 0] = x[v];
        hin[v * 3 + 1] = pos[v * 3 + 0];
        hin[v * 3 + 2] = pos[v * 3 + 1];
    }
}

// ---------------- per-dst degree ----------------
__global__ void k_deg(const int* __restrict__ dst, float* __restrict__ deg, int E) {
    int e = blockIdx.x * blockDim.x + threadIdx.x;
    if (e < E) atomicAdd(&deg[dst[e]], 1.0f);
}

// ---------------- fused edge-message GEMM via f32 WMMA + atomic scatter ----------------
// One wave computes a 16-edge x (NT*16)-outchannel tile. The A-fragment
//   A[m][k] = basis_{k/Cin}(edge m) * hin[src[m]][k%Cin]
// is built once per K-step and reused across NT column tiles (NT WMMAs back-to-back).
template <int NT>
__global__ void k_spline_edge(const int* __restrict__ src, const int* __restrict__ dst,
                              const float* __restrict__ hin, const float* __restrict__ pos,
                              const float* __restrict__ W, float* __restrict__ agg,
                              int E, int Cin, int Cout, float inv2mv) {
    __shared__ float s_xs[WPB][16 * MAX_CIN];   // staged x[src] rows
    __shared__ float s_basis[WPB][16 * 8];      // 8 trilinear corner weights per edge
    __shared__ int   s_dst[WPB][16];
    __shared__ short s_ks[MAX_K];               // k -> corner s
    __shared__ short s_kc[MAX_K];               // k -> channel c

    const int K = 8 * Cin;
    for (int k = threadIdx.x; k < K; k += blockDim.x) {
        int sC = k / Cin;
        s_ks[k] = (short)sC;
        s_kc[k] = (short)(k - sC * Cin);
    }

    const int wave  = threadIdx.x >> 5;
    const int lane  = threadIdx.x & 31;
    const int eBase = (blockIdx.x * WPB + wave) * 16;

    if (lane < 16) {
        int eI = eBase + lane;
        int sN = src[eI], dN = dst[eI];
        s_dst[wave][lane] = dN;
        float p0 = (pos[dN * 3 + 0] - pos[sN * 3 + 0]) * inv2mv + 0.5f;
        float p1 = (pos[dN * 3 + 1] - pos[sN * 3 + 1]) * inv2mv + 0.5f;
        float p2 = (pos[dN * 3 + 2] - pos[sN * 3 + 2]) * inv2mv + 0.5f;
        p0 = fminf(fmaxf(p0, 0.0f), 1.0f);
        p1 = fminf(fmaxf(p1, 0.0f), 1.0f);
        p2 = fminf(fmaxf(p2, 0.0f), 1.0f);
        float q0 = 1.0f - p0, q1 = 1.0f - p1, q2 = 1.0f - p2;
#pragma unroll
        for (int s8 = 0; s8 < 8; s8++) {
            s_basis[wave][lane * 8 + s8] =
                ((s8 & 1) ? p0 : q0) * ((s8 & 2) ? p1 : q1) * ((s8 & 4) ? p2 : q2);
        }
    }
    // stage the 16 source-node feature rows (uniform src per row -> scalar load)
    for (int m2 = 0; m2 < 16; m2++) {
        long sN = src[eBase + m2];
        for (int c = lane; c < Cin; c += 32)
            s_xs[wave][m2 * Cin + c] = hin[sN * Cin + c];
    }
    __syncthreads();

    const int hi = lane >> 4;                    // lane half selects K pair
    const int m  = lane & 15;                    // edge row (A) / out column (B, D)
    const int n0 = blockIdx.y * (NT * 16) + m;   // first column this lane covers
    const float* Wn  = W + n0;
    const float* bas = &s_basis[wave][m * 8];
    const float* xsr = &s_xs[wave][m * Cin];

    v8f acc[NT];
#pragma unroll
    for (int t = 0; t < NT; t++) acc[t] = v8f{};

    for (int kk = 0; kk < K; kk += 4) {
        v2f a = {};
        long krow[2];
#pragma unroll
        for (int v = 0; v < 2; v++) {
            int k = kk + 2 * hi + v;             // A: M=lane%16, K = v + 2*(lane/16)
            a[v] = bas[s_ks[k]] * xsr[s_kc[k]];
            krow[v] = (long)k * Cout;
        }
#pragma unroll
        for (int t = 0; t < NT; t++) {
            v2f b = {};
#pragma unroll
            for (int v = 0; v < 2; v++)
                b[v] = Wn[krow[v] + t * 16];     // B: K = v + 2*(lane/16), N = lane%16
            acc[t] = __builtin_amdgcn_wmma_f32_16x16x4_f32(
                false, a, false, b, (short)0, acc[t], false, false);
        }
    }

    // D layout: lane l, vgpr v -> row v + 8*(l/16), col l%16
#pragma unroll
    for (int t = 0; t < NT; t++) {
#pragma unroll
        for (int v = 0; v < 8; v++) {
            int row = v + 8 * hi;
            atomicAdd(&agg[(long)s_dst[wave][row] * Cout + n0 + t * 16], acc[t][v]);
        }
    }
}

// ---------------- out = relu(agg/max(deg,1) + hin @ Wr + b) ----------------
__global__ void k_node_update(const float* __restrict__ agg, const float* __restrict__ deg,
                              const float* __restrict__ hin, const float* __restrict__ R,
                              const float* __restrict__ bias, float* __restrict__ hout,
                              int V, int Cin, int Cout) {
    long i = (long)blockIdx.x * blockDim.x + threadIdx.x;
    if (i >= (long)V * Cout) return;
    int v = (int)(i / Cout);
    int j = (int)(i - (long)v * Cout);
    float acc = agg[i] / fmaxf(deg[v], 1.0f) + bias[j];
    for (int c = 0; c < Cin; c++)
        acc += hin[(long)v * Cin + c] * R[(long)c * Cout + j];
    hout[i] = fmaxf(acc, 0.0f);
}

// ---------------- voxel pool: scatter ----------------
__global__ void k_pool_scatter(const float* __restrict__ h, const float* __restrict__ pos,
                               const int* __restrict__ batch_in, int batch_div,
                               int V, int C, int nx, int ny, int nt, int Cnext,
                               float* __restrict__ hin_next, float* __restrict__ psum,
                               float* __restrict__ cnt, int is_max) {
    int v = blockIdx.x * blockDim.x + threadIdx.x;
    if (v >= V) return;
    int b = batch_in ? batch_in[v] : (v / batch_div);
    float p0 = pos[v * 3 + 0], p1 = pos[v * 3 + 1], p2 = pos[v * 3 + 2];
    int ix = min(max((int)floorf(p0 * (float)nx), 0), nx - 1);
    int iy = min(max((int)floorf(p1 * (float)ny), 0), ny - 1);
    int it = min(max((int)floorf(p2 * (float)nt), 0), nt - 1);
    long cl = (((long)b * nx + ix) * ny + iy) * nt + it;
    atomicAdd(&cnt[cl], 1.0f);
    atomicAdd(&psum[cl * 3 + 0], p0);
    atomicAdd(&psum[cl * 3 + 1], p1);
    atomicAdd(&psum[cl * 3 + 2], p2);
    float* base = &hin_next[cl * Cnext];
    for (int j = 0; j < C; j++) {
        float val = h[(long)v * C + j];  // post-ReLU => val >= 0, so uint-max == float-max
        if (is_max) atomicMax((unsigned int*)&base[j], __float_as_uint(val));
        else        atomicAdd(&base[j], val);
    }
}

// ---------------- voxel pool: finalize (pos mean, optional feature mean, pos.xy concat) --
__global__ void k_pool_finalize(const float* __restrict__ psum, const float* __restrict__ cnt,
                                float* __restrict__ pos_next, float* __restrict__ hin_next,
                                int V2, int Cout, int Cnext, int is_mean) {
    int v = blockIdx.x * blockDim.x + threadIdx.x;
    if (v >= V2) return;
    float c = fmaxf(cnt[v], 1.0f);
    float px = psum[v * 3 + 0] / c, py = psum[v * 3 + 1] / c, pz = psum[v * 3 + 2] / c;
    pos_next[v * 3 + 0] = px; pos_next[v * 3 + 1] = py; pos_next[v * 3 + 2] = pz;
    float* row = &hin_next[(long)v * Cnext];
    if (is_mean) {
        for (int j = 0; j < Cout; j++) row[j] /= c;
    }
    row[Cout + 0] = px;
    row[Cout + 1] = py;
}

extern "C" void kernel_launch(void* const* d_in, const int* in_sizes, int n_in,
                              void* d_out, int out_size, void* d_ws, size_t ws_size,
                              hipStream_t stream) {
    const float* x      = (const float*)d_in[0];
    const float* pos0   = (const float*)d_in[1];
    const int*   batch0 = (const int*)d_in[2];
    const int* eptr[5];
    for (int i = 0; i < 5; i++) eptr[i] = (const int*)d_in[3 + i];
    const float *Wk[5], *Rr[5], *Bb[5];
    for (int i = 0; i < 5; i++) {
        Wk[i] = (const float*)d_in[8 + 3 * i];
        Rr[i] = (const float*)d_in[9 + 3 * i];
        Bb[i] = (const float*)d_in[10 + 3 * i];
    }

    static const int   Vs[5] = {300000, 98304, 12288, 1536, 192};
    static const int   Es[5] = {3600000, 786432, 196608, 24576, 3072};
    static const int   Ci[5] = {3, 18, 66, 130, 130};
    static const int   Co[5] = {16, 64, 128, 128, 128};
    static const float Mv[5] = {0.025f, 0.05f, 2.0f / 24.0f, 2.0f / 12.0f, 2.0f / 6.0f};
    static const int GX[4] = {64, 32, 16, 8}, GY[4] = {48, 24, 12, 6}, GT[4] = {8, 4, 2, 1};

    float* ws = (float*)d_ws;
    size_t off = 0;
    auto carve = [&](size_t n) { float* p = ws + off; off += n; return p; };
    float* hinA = carve(2000000);
    float* hinB = carve(2000000);
    float* hbuf = carve(6400000);
    float* agg  = carve(6400000);
    float* deg  = carve(300000);
    float* posA = carve(98304 * 3);
    float* posB = carve(98304 * 3);
    float* cnt  = carve(98304);
    float* psum = carve(98304 * 3);
    (void)ws_size; (void)in_sizes; (void)n_in; (void)out_size;

    k_build_hin0<<<(Vs[0] + 255) / 256, 256, 0, stream>>>(x, pos0, hinA, Vs[0]);

    float* hinCur = hinA;  float* hinNext = hinB;
    const float* posCur = pos0;  float* posNext = posA;
    const int* batchPtr = batch0;  int batchDiv = 1;

    for (int l = 0; l < 5; l++) {
        int V = Vs[l], E = Es[l], Cin = Ci[l], Cout = Co[l];
        const int* src = eptr[l];
        const int* dst = eptr[l] + E;

        hipMemsetAsync(agg, 0, (size_t)V * Cout * sizeof(float), stream);
        hipMemsetAsync(deg, 0, (size_t)V * sizeof(float), stream);
        k_deg<<<(E + 255) / 256, 256, 0, stream>>>(dst, deg, E);

        float i2m = 1.0f / (2.0f * Mv[l]);
        if (Cout == 16) {
            dim3 g(E / (WPB * 16), 1);
            k_spline_edge<1><<<g, WPB * 32, 0, stream>>>(src, dst, hinCur, posCur, Wk[l],
                                                         agg, E, Cin, Cout, i2m);
        } else if (Cout == 64) {
            dim3 g(E / (WPB * 16), 1);
            k_spline_edge<4><<<g, WPB * 32, 0, stream>>>(src, dst, hinCur, posCur, Wk[l],
                                                         agg, E, Cin, Cout, i2m);
        } else {  // Cout == 128
            dim3 g(E / (WPB * 16), 1);
            k_spline_edge<8><<<g, WPB * 32, 0, stream>>>(src, dst, hinCur, posCur, Wk[l],
                                                         agg, E, Cin, Cout, i2m);
        }

        long tot = (long)V * Cout;
        k_node_update<<<(unsigned)((tot + 255) / 256), 256, 0, stream>>>(
            agg, deg, hinCur, Rr[l], Bb[l], hbuf, V, Cin, Cout);

        if (l == 3)  // out3 = features after layer 4 (pre-pool)
            hipMemcpyAsync(d_out, hbuf, (size_t)1536 * 128 * sizeof(float),
                           hipMemcpyDeviceToDevice, stream);
        if (l == 4)
            hipMemcpyAsync((float*)d_out + 196608, hbuf, (size_t)192 * 128 * sizeof(float),
                           hipMemcpyDeviceToDevice, stream);

        if (l < 4) {
            int V2 = Vs[l + 1], Cn = Ci[l + 1];
            hipMemsetAsync(cnt, 0, (size_t)V2 * sizeof(float), stream);
            hipMemsetAsync(psum, 0, (size_t)V2 * 3 * sizeof(float), stream);
            hipMemsetAsync(hinNext, 0, (size_t)V2 * Cn * sizeof(float), stream);
            k_pool_scatter<<<(V + 255) / 256, 256, 0, stream>>>(
                hbuf, posCur, batchPtr, batchDiv, V, Cout, GX[l], GY[l], GT[l], Cn,
                hinNext, psum, cnt, (l < 3) ? 1 : 0);
            k_pool_finalize<<<(V2 + 255) / 256, 256, 0, stream>>>(
                psum, cnt, posNext, hinNext, V2, Cout, Cn, (l == 3) ? 1 : 0);

            posCur = posNext;
            posNext = (posNext == posA) ? posB : posA;
            float* t = hinCur; hinCur = hinNext; hinNext = t;
            batchPtr = nullptr;
            batchDiv = GX[l] * GY[l] * GT[l];
        }
    }
}